// CARAFE_5446018531573
// MI455X (gfx1250) — compile-verified
//
#include <hip/hip_runtime.h>
#include <hip/hip_bf16.h>

// CARAFE upsampling as per-pixel GEMM on CDNA5 WMMA (fp32 path).
//   out[n,c,2h+p,2w+q] = sum_{t<25} masks[n,t,2h+p,2w+q] * feat[n,c,h+t/5-2,w+t%5-2]
// Per dense 16x16 WMMA D-tile: M = 16 channels, N = 4 pixels x 4 subpixels,
// K = 4 pixels x 28 taps (block-diagonal B), using V_WMMA_F32_16X16X4_F32.
// Staging uses GLOBAL_LOAD_ASYNC_TO_LDS_B32 (ASYNCcnt) where available.

typedef __attribute__((ext_vector_type(2))) float v2f;
typedef __attribute__((ext_vector_type(8))) float v8f;

#define CN 4
#define CC 256
#define CH 64
#define CW 64
// workgroup tile: 4 rows x 8 cols of low-res pixels, 128 channels (8 waves x 16)
#define TROWS 4
#define TCOLS 8
#define WROWS 8    // TROWS + 4 halo
#define WCOLS 12   // TCOLS + 4 halo
#define CHB   128  // channels per workgroup
#define FSTR  97   // odd channel stride -> conflict-free 16-lane channel reads
#define NSLOT 7    // 28 tap slots (25 real + 3 dummy with B==0)
#define MPIX  112  // NSLOT*16 floats per pixel block in sM
#define NPIXB 33   // 32 pixels + 1 zero block (block-diagonal zeros)

#if __has_builtin(__builtin_amdgcn_global_load_async_to_lds_b32)
#define HAVE_ASYNC_LDS 1
#else
#define HAVE_ASYNC_LDS 0
#endif

#if HAVE_ASYNC_LDS
typedef __attribute__((address_space(1))) int g_i32;
typedef __attribute__((address_space(3))) int l_i32;
// global generic ptr -> AS1, LDS generic ptr -> AS3 (low 32 bits = LDS offset)
__device__ __forceinline__ void async_g2l_b32(const float* g, const float* l) {
    __builtin_amdgcn_global_load_async_to_lds_b32(
        (g_i32*)(uintptr_t)g,
        (l_i32*)(unsigned)(uintptr_t)l,
        0, 0);
}
__device__ __forceinline__ void wait_async0() {
#if __has_builtin(__builtin_amdgcn_s_wait_asynccnt)
    __builtin_amdgcn_s_wait_asynccnt(0);
#else
    asm volatile("s_wait_asynccnt 0x0" ::: "memory");
#endif
}
#endif

__global__ __launch_bounds__(256) void carafe_wmma_kernel(
    const float* __restrict__ feat,   // [4,256,64,64]
    const float* __restrict__ masks,  // [4,25,128,128]
    float* __restrict__ out)          // [4,256,128,128]
{
    __shared__ float sF[CHB * FSTR];   // 49664 B
    __shared__ float sM[NPIXB * MPIX]; // 14784 B  (total 64448 B LDS)

    const int tid   = threadIdx.x;
    const int wave  = tid >> 5;
    const int lane  = tid & 31;

    const int wb    = blockIdx.x * TCOLS;       // low-res col base
    const int hb    = blockIdx.y * TROWS;       // low-res row base
    const int chalf = blockIdx.z & 1;
    const int n     = blockIdx.z >> 1;

#if HAVE_ASYNC_LDS
    // workgroup-uniform: is the 12x8 halo window fully in-bounds?
    const bool interior = (hb != 0) && (hb != CH - TROWS) &&
                          (wb != 0) && (wb != CW - TCOLS);
#else
    const bool interior = false;
#endif

    // ---- phase 1: zero sM (dummy taps + zero pixel block); edge path also
    //      stages features with zero-halo blending through VGPRs ----
    for (int idx = tid; idx < NPIXB * MPIX; idx += 256) sM[idx] = 0.0f;
    if (!interior) {
        for (int idx = tid; idx < CHB * WROWS * WCOLS; idx += 256) {
            int c   = idx / (WROWS * WCOLS);
            int rem = idx - c * (WROWS * WCOLS);
            int row = rem / WCOLS;
            int col = rem - row * WCOLS;
            int gr  = hb - 2 + row;
            int gc  = wb - 2 + col;
            float v = 0.0f;
            if ((unsigned)gr < (unsigned)CH && (unsigned)gc < (unsigned)CW)
                v = feat[(((size_t)n * CC + chalf * CHB + c) * CH + gr) * CW + gc];
            sF[c * FSTR + rem] = v;
        }
    }
    __syncthreads();

    // ---- phase 2: async global->LDS staging (per-lane LDS scatter) ----
#if HAVE_ASYNC_LDS
    if (interior) {
        for (int idx = tid; idx < CHB * WROWS * WCOLS; idx += 256) {
            int c   = idx / (WROWS * WCOLS);
            int rem = idx - c * (WROWS * WCOLS);
            int row = rem / WCOLS;
            int col = rem - row * WCOLS;
            async_g2l_b32(
                &feat[(((size_t)n * CC + chalf * CHB + c) * CH + (hb - 2 + row)) * CW
                      + (wb - 2 + col)],
                &sF[c * FSTR + rem]);
        }
    }
#endif
    // masks: layout sM[pix][slot][khalf][sub][tap-pair]; always in-bounds
    for (int idx = tid; idx < 25 * 128; idx += 256) {
        int t    = idx >> 7;          // tap 0..24
        int rem  = idx & 127;
        int hi_l = rem >> 4;          // 0..7 hi-res row in tile
        int wi_l = rem & 15;          // 0..15 hi-res col in tile
        int p    = hi_l & 1, rowT = hi_l >> 1;
        int q    = wi_l & 1, colT = wi_l >> 1;
        int pix  = rowT * TCOLS + colT;
        int sub  = p * 2 + q;
        int slot = t >> 2;            // t = slot*4 + kh*2 + tp
        int kh   = (t >> 1) & 1;
        int tp   = t & 1;
        const float* gsrc = &masks[(((size_t)n * 25 + t) * (CH * 2) + (2 * hb + hi_l))
                                   * (CW * 2) + (2 * wb + wi_l)];
        float* ldst = &sM[pix * MPIX + slot * 16 + kh * 8 + sub * 2 + tp];
#if HAVE_ASYNC_LDS
        async_g2l_b32(gsrc, ldst);
#else
        *ldst = *gsrc;
#endif
    }
#if HAVE_ASYNC_LDS
    wait_async0();
#endif
    __syncthreads();

    // ---- per-lane invariants (ISA 16x16x4 f32 operand layouts) ----
    const int n16   = lane & 15;        // A: channel row M ; B/D: column N
    const int kh    = lane >> 4;        // lo lanes: K 0,1 ; hi lanes: K 2,3
    const int jlane = n16 >> 2;         // pixel-in-quad of this column
    const int sub   = n16 & 3;          // subpixel of this column
    const int khsub = kh * 8 + sub * 2;
    const int kh2   = kh * 2;

    // A-operand LDS offsets per slot (dummy taps -> 0, B is 0 there anyway)
    int offA0[NSLOT], offA1[NSLOT];
#pragma unroll
    for (int s = 0; s < NSLOT; ++s) {
        int t0 = s * 4 + kh2, t1 = t0 + 1;
        offA0[s] = (t0 < 25) ? (t0 / 5) * WCOLS + (t0 % 5) : 0;
        offA1[s] = (t1 < 25) ? (t1 / 5) * WCOLS + (t1 % 5) : 0;
    }

    const int cbase = chalf * CHB + wave * 16;          // global channel base
    const int afix  = (wave * 16 + n16) * FSTR;         // per-lane channel block

    // ---- 8 quads per wave: one dense 16x16 D tile each ----
    for (int qd = 0; qd < 8; ++qd) {
        const int r  = qd >> 1;        // tile row 0..3
        const int qc = qd & 1;         // quad col 0..1 (4 pixels each)
        v8f acc = {};

        const int abase = afix + r * WCOLS + qc * 4;
#pragma unroll
        for (int jseg = 0; jseg < 4; ++jseg) {
            const int apix = abase + jseg;
            const int pixq = r * TCOLS + qc * 4 + jseg;
            // block-diagonal B: inactive columns read the zeroed pixel block 32
            const int mb = ((jlane == jseg) ? pixq : 32) * MPIX + khsub;
#pragma unroll
            for (int s = 0; s < NSLOT; ++s) {
                v2f a;
                a.x = sF[apix + offA0[s]];
                a.y = sF[apix + offA1[s]];
                v2f b = *(const v2f*)(sM + mb + s * 16);   // ds_load_b64, imm off
                acc = __builtin_amdgcn_wmma_f32_16x16x4_f32(
                    false, a, false, b, (short)0, acc, false, false);
            }
        }

        // ---- store dense D tile: lane col -> (pixel, subpixel) ----
        const int wp = wb + qc * 4 + jlane;
        const int oy = 2 * (hb + r) + (sub >> 1);
        const int ox = 2 * wp + (sub & 1);
        size_t obase = (((size_t)n * CC + cbase + kh * 8) * (CH * 2) + oy) * (CW * 2) + ox;
#pragma unroll
        for (int r8 = 0; r8 < 8; ++r8)
            out[obase + (size_t)r8 * (CH * 2) * (CW * 2)] = acc[r8];
    }
}

extern "C" void kernel_launch(void* const* d_in, const int* in_sizes, int n_in,
                              void* d_out, int out_size, void* d_ws, size_t ws_size,
                              hipStream_t stream) {
    const float* feat  = (const float*)d_in[0];  // [4,256,64,64]
    const float* masks = (const float*)d_in[1];  // [4,25,128,128]
    float* out = (float*)d_out;                  // [4,256,128,128]
    dim3 grid(CW / TCOLS, CH / TROWS, CN * 2);   // (8,16,8) = 1024 workgroups
    dim3 block(256);
    carafe_wmma_kernel<<<grid, block, 0, stream>>>(feat, masks, out);
}